// VGGBlock_83708912599749
// MI455X (gfx1250) — compile-verified
//
#include <hip/hip_runtime.h>

typedef __attribute__((ext_vector_type(2))) float v2f;
typedef __attribute__((ext_vector_type(8))) float v8f;

#define BATCH 8
#define CH    256
#define HH    128
#define WW    128
#define NTHRESH 0.1f

#define XPAD   18      // channel-dim padding: even (keeps b64 align), spreads LDS banks
#define XSLOTS 132     // row slots: w = -1..130 stored at slot w+1

// Async global->LDS path (CDNA5 GLOBAL_LOAD_ASYNC_TO_LDS_B32), guarded so the
// build stays green if this toolchain doesn't declare the builtins.
#if defined(__has_builtin)
#if __has_builtin(__builtin_amdgcn_global_load_async_to_lds_b32) && \
    __has_builtin(__builtin_amdgcn_s_wait_asynccnt)
#define USE_ASYNC_LDS 1
#endif
#endif

#if defined(USE_ASYNC_LDS)
__device__ __forceinline__ void async_fill_b32(const float* g, float* l) {
  // builtin signature (per hipcc diagnostic): (int*, int*, imm offset, imm cpol)
  __builtin_amdgcn_global_load_async_to_lds_b32((int*)g, (int*)l, 0, 0);
}
#endif

// ---------------------------------------------------------------------------
// Implicit-GEMM 3x3 SAME conv via V_WMMA_F32_16X16X4_F32.
// Block = 256 threads (8 waves). Block computes out[b][oc0:oc0+32][h][0:128].
// Each wave: 16 consecutive pixels (M tile) x 32 output channels (2 N tiles,
// A fragment reused for 2 WMMAs). K = 256 in-channels x 9 taps, 4 at a time.
// ---------------------------------------------------------------------------
__global__ __launch_bounds__(256) void nanconv3x3_wmma(
    const float* __restrict__ act,    // pre-masked activation, NCHW
    const float* __restrict__ wgt,    // OIHW (256,256,3,3)
    const float* __restrict__ bias,   // (256)
    float* __restrict__ out)          // NCHW
{
  __shared__ __align__(16) float xlds[3 * XSLOTS * XPAD]; // [row r][slot][ci]
  __shared__ __align__(16) float wlds[9 * 32 * XPAD];     // [tap][n(32)][ci]

  const int tid   = threadIdx.x;
  const int wave  = tid >> 5;
  const int lane  = tid & 31;
  const int half  = lane >> 4;   // K-half selector for A/B fragments
  const int idx16 = lane & 15;   // A: M row / B,D: N col

  const int oc0 = blockIdx.x * 32;
  const int h   = blockIdx.y;
  const int b   = blockIdx.z;

  const float* actb = act + (size_t)b * CH * HH * WW;

  v8f acc0 = {};
  v8f acc1 = {};

  #pragma unroll 1
  for (int cc = 0; cc < CH; cc += 16) {
    // ---- stage 16 input channels x 3 halo rows x 130 cols into LDS ----
    for (int i = tid; i < 3 * XSLOTS * 16; i += 256) {
      int slot = i % XSLOTS;
      int row  = i / XSLOTS;        // 0..47 = ci*3 + r
      int ci   = row / 3;
      int r    = row - ci * 3;
      int wg   = slot - 1;          // global col
      int hg   = h + r - 1;         // global row
      float* dst = &xlds[(r * XSLOTS + slot) * XPAD + ci];
      bool inb = ((unsigned)wg < (unsigned)WW) && ((unsigned)hg < (unsigned)HH);
#if defined(USE_ASYNC_LDS)
      if (inb) async_fill_b32(&actb[((size_t)(cc + ci) * HH + hg) * WW + wg], dst);
      else     *dst = 0.0f;
#else
      float v = 0.0f;
      if (inb) v = actb[((size_t)(cc + ci) * HH + hg) * WW + wg];
      *dst = v;
#endif
    }
    // ---- stage weights as wlds[tap][n][ci] (ci pairs contiguous -> b64) ----
    for (int i = tid; i < 9 * 32 * 16; i += 256) {
      int tap = i % 9;              // contiguous global reads per (n,ci) run
      int c2  = i / 9;
      int ci  = c2 & 15;
      int n   = c2 >> 4;            // 0..31
      const float* src = &wgt[(((size_t)(oc0 + n)) * CH + (cc + ci)) * 9 + tap];
      float* dst = &wlds[(tap * 32 + n) * XPAD + ci];
#if defined(USE_ASYNC_LDS)
      async_fill_b32(src, dst);
#else
      *dst = *src;
#endif
    }
#if defined(USE_ASYNC_LDS)
    __builtin_amdgcn_s_wait_asynccnt(0);
#endif
    __syncthreads();

    // ---- 9 taps x 4 K-steps; each A fragment feeds 2 WMMAs (N blocking) ----
    const int pix = wave * 16 + idx16;       // A-fragment row (pixel)
    #pragma unroll
    for (int tap = 0; tap < 9; ++tap) {
      const int kh = tap / 3, kw = tap % 3;
      const float* xrow  = &xlds[(kh * XSLOTS + pix + kw) * XPAD];
      const float* wrow0 = &wlds[(tap * 32 + idx16) * XPAD];
      const float* wrow1 = wrow0 + 16 * XPAD;
      #pragma unroll
      for (int k4 = 0; k4 < 4; ++k4) {
        const int ci = k4 * 4 + half * 2;    // this lane's 2 K values
        v2f afrag = *(const v2f*)(xrow + ci);
        v2f bf0   = *(const v2f*)(wrow0 + ci);
        v2f bf1   = *(const v2f*)(wrow1 + ci);
        acc0 = __builtin_amdgcn_wmma_f32_16x16x4_f32(
            false, afrag, false, bf0, (short)0, acc0, false, false);
        acc1 = __builtin_amdgcn_wmma_f32_16x16x4_f32(
            false, afrag, false, bf1, (short)0, acc1, false, false);
      }
    }
    __syncthreads();
  }

  // ---- epilogue: lane holds D[8*half + r][idx16], r = 0..7 ----
  const int oc_a = oc0 + idx16;
  const int oc_b = oc_a + 16;
  const float bva = bias[oc_a];
  const float bvb = bias[oc_b];
  const size_t base = (size_t)b * CH * HH * WW + (size_t)h * WW
                    + wave * 16 + half * 8;
  float* oa = out + base + (size_t)oc_a * (HH * WW);
  float* ob = out + base + (size_t)oc_b * (HH * WW);
  #pragma unroll
  for (int r = 0; r < 8; ++r) {
    oa[r] = acc0[r] + bva;
    ob[r] = acc1[r] + bvb;
  }
}

// ---------------------------------------------------------------------------
// Fused elementwise stage: m = max(p,q); optionally store m;
// t = mask(prelu(m)) — the next conv's (pre-masked) input.
// ---------------------------------------------------------------------------
__device__ __forceinline__ float prelu_mask(float v, float a) {
  float u = (v >= 0.0f) ? v : a * v;
  return (fabsf(u) >= NTHRESH) ? u : 0.0f;
}

__global__ __launch_bounds__(256) void fuse_max_prelu_mask(
    const float4* __restrict__ p, const float4* __restrict__ q,
    float4* __restrict__ t, float4* __restrict__ xmax,
    const float* __restrict__ a_ptr, int n4)
{
  int i = blockIdx.x * 256 + threadIdx.x;
  if (i >= n4) return;
  const float a = a_ptr[0];
  float4 pv = p[i], qv = q[i], mv, tv;
  mv.x = fmaxf(pv.x, qv.x);
  mv.y = fmaxf(pv.y, qv.y);
  mv.z = fmaxf(pv.z, qv.z);
  mv.w = fmaxf(pv.w, qv.w);
  if (xmax) xmax[i] = mv;
  tv.x = prelu_mask(mv.x, a);
  tv.y = prelu_mask(mv.y, a);
  tv.z = prelu_mask(mv.z, a);
  tv.w = prelu_mask(mv.w, a);
  t[i] = tv;
}

// ---------------------------------------------------------------------------
extern "C" void kernel_launch(void* const* d_in, const int* in_sizes, int n_in,
                              void* d_out, int out_size, void* d_ws, size_t ws_size,
                              hipStream_t stream)
{
  const float* x  = (const float*)d_in[0];
  const float* w1 = (const float*)d_in[1];
  const float* b1 = (const float*)d_in[2];
  const float* w2 = (const float*)d_in[3];
  const float* b2 = (const float*)d_in[4];
  const float* w3 = (const float*)d_in[5];
  const float* b3 = (const float*)d_in[6];
  const float* pa = (const float*)d_in[7];

  const size_t N = (size_t)BATCH * CH * HH * WW;   // 33,554,432
  float* B0 = (float*)d_ws;      // conv input (masked activation)
  float* B1 = B0 + N;            // conv output
  float* B2 = B1 + N;            // running max carried to stage 3

  const int n4 = (int)(N / 4);
  dim3 eg((n4 + 255) / 256), eb(256);
  dim3 cg(CH / 32, HH, BATCH), cb(256);

  // t0 = mask(prelu(x))
  fuse_max_prelu_mask<<<eg, eb, 0, stream>>>((const float4*)x, (const float4*)x,
                                             (float4*)B0, nullptr, pa, n4);
  // x0 = conv1(t0) + b1
  nanconv3x3_wmma<<<cg, cb, 0, stream>>>(B0, w1, b1, B1);
  // x1_max = max(x, x0); t1 = mask(prelu(x1_max))
  fuse_max_prelu_mask<<<eg, eb, 0, stream>>>((const float4*)x, (const float4*)B1,
                                             (float4*)B0, (float4*)B2, pa, n4);
  // x1c = conv2(t1) + b2
  nanconv3x3_wmma<<<cg, cb, 0, stream>>>(B0, w2, b2, B1);
  // x2_max = max(x1c, x1_max); t2 = mask(prelu(x2_max))
  fuse_max_prelu_mask<<<eg, eb, 0, stream>>>((const float4*)B1, (const float4*)B2,
                                             (float4*)B0, nullptr, pa, n4);
  // out = conv3(t2) + b3
  nanconv3x3_wmma<<<cg, cb, 0, stream>>>(B0, w3, b3, (float*)d_out);
}